// SinkhornOTLoss_46815143527069
// MI455X (gfx1250) — compile-verified
//
#include <hip/hip_runtime.h>
#include <hip/hip_bf16.h>

// Sinkhorn OT loss for MI455X (gfx1250), compile-only target.
// n=2000 topics, m=20000 vocab.  K = exp(-20*M) is 160 MB -> fits in the
// 192 MB global L2, so we materialize K into workspace once and stream all
// ~200 matvec passes from L2.  K^T u is done with V_WMMA_F32_16X16X4_F32
// (coalesced B-operand loads along K's rows); K v is a float4 streaming
// block reduction.  Data-dependent loop exit (cpt=1/51 convergence checks)
// is handled with a device-side done flag so the enqueued work is fixed.

typedef __attribute__((ext_vector_type(2))) float v2f;
typedef __attribute__((ext_vector_type(8))) float v8f;

#define N_TOPICS 2000
#define M_VOCAB  20000
#define ALPHA    20.0f
#define WLOSS    100.0f
#define EPS_S    1e-16f
#define THR      0.005f

// ---------------------------------------------------------------- init
__global__ void sk_init(float* __restrict__ u, int* __restrict__ done,
                        float* __restrict__ err, float* __restrict__ out) {
  int i = blockIdx.x * blockDim.x + threadIdx.x;
  if (i < N_TOPICS) u[i] = 1.0f / (float)N_TOPICS;
  if (i == 0) { *done = 0; *err = 1.0f; *out = 0.0f; }
}

// ---------------------------------------------------------------- K = exp(-20 M)
__global__ void sk_buildK(const float* __restrict__ M, float* __restrict__ K) {
  long long idx = ((long long)blockIdx.x * blockDim.x + threadIdx.x) * 4;
  const long long total = (long long)N_TOPICS * M_VOCAB;   // multiple of 4
  if (idx + 3 < total) {
    float4 mv = *(const float4*)(M + idx);
    float4 kv;
    kv.x = __expf(-ALPHA * mv.x);
    kv.y = __expf(-ALPHA * mv.y);
    kv.z = __expf(-ALPHA * mv.z);
    kv.w = __expf(-ALPHA * mv.w);
    *(float4*)(K + idx) = kv;
  }
}

// ---------------------------------------------------------------- WMMA col-sum tile
// One wave owns 16 columns [j0, j0+16) and sweeps all rows.
// t[j] = sum_i u[i] * K[i][j], via D(16x16) += A(16x4) x B(4x16) where
// A[m][k] = u[ib+k] (replicated over rows m), B[k][n] = K[ib+k][j0+n].
// f32 A 16x4 layout: lanes 0-15 hold K=0(v0),K=1(v1); lanes 16-31 K=2,K=3.
template<int USEK>
__device__ __forceinline__ float sk_colsum_tile16(const float* __restrict__ Kp,
                                                  const float* __restrict__ Mp,
                                                  const float* __restrict__ u,
                                                  int j0) {
  const int lane = threadIdx.x & 31;
  const int hi   = (lane >= 16) ? 1 : 0;
  const int jc   = j0 + (lane & 15);
  const int k0   = hi ? 2 : 0;
  const int k1   = hi ? 3 : 1;

  v8f acc0 = {}; v8f acc1 = {}; v8f acc2 = {}; v8f acc3 = {};

  for (int i0 = 0; i0 < N_TOPICS; i0 += 16) {
#pragma unroll
    for (int s = 0; s < 4; ++s) {
      const int ib = i0 + 4 * s;
      v2f a; a.x = u[ib + k0]; a.y = u[ib + k1];
      v2f b;
      const long long r0 = (long long)(ib + k0) * M_VOCAB + jc;
      const long long r1 = (long long)(ib + k1) * M_VOCAB + jc;
      if (USEK) { b.x = Kp[r0];                    b.y = Kp[r1]; }
      else      { b.x = __expf(-ALPHA * Mp[r0]);   b.y = __expf(-ALPHA * Mp[r1]); }
#if __has_builtin(__builtin_amdgcn_wmma_f32_16x16x4_f32)
      v8f* accp = (s == 0) ? &acc0 : (s == 1) ? &acc1 : (s == 2) ? &acc2 : &acc3;
      *accp = __builtin_amdgcn_wmma_f32_16x16x4_f32(
          /*neg_a=*/false, a, /*neg_b=*/false, b,
          /*c_mod=*/(short)0, *accp, /*reuse_a=*/false, /*reuse_b=*/false);
#else
      // VALU fallback: each half-wave handles its two K rows, combined below.
      float p = a.x * b.x + a.y * b.y;
      if (s == 0) acc0[0] += p; else if (s == 1) acc1[0] += p;
      else if (s == 2) acc2[0] += p; else acc3[0] += p;
#endif
    }
  }
  v8f t = acc0 + acc1 + acc2 + acc3;
  float tt = t[0];
#if !__has_builtin(__builtin_amdgcn_wmma_f32_16x16x4_f32)
  tt += __shfl_xor(tt, 16, 32);   // merge K={0,1} and K={2,3} halves
#endif
  return tt;   // valid in lanes 0-15: partial col sums for j0+lane
}

// v[j] = (1/m) / (K^T u + eps)
template<int USEK>
__global__ void __launch_bounds__(32)
sk_colsum_v(const float* __restrict__ Kp, const float* __restrict__ Mp,
            const float* __restrict__ u, float* __restrict__ v,
            const int* __restrict__ done) {
  if (*done) return;
  const int j0 = blockIdx.x * 16;
  float t = sk_colsum_tile16<USEK>(Kp, Mp, u, j0);
  if ((threadIdx.x & 31) < 16)
    v[j0 + (threadIdx.x & 15)] = (1.0f / (float)M_VOCAB) / (t + EPS_S);
}

// err += |v_j * (K^T u)_j - 1/m|   (with post-update u,v)
template<int USEK>
__global__ void __launch_bounds__(32)
sk_colsum_err(const float* __restrict__ Kp, const float* __restrict__ Mp,
              const float* __restrict__ u, const float* __restrict__ v,
              float* __restrict__ err, const int* __restrict__ done) {
  if (*done) return;
  const int j0 = blockIdx.x * 16;
  float t = sk_colsum_tile16<USEK>(Kp, Mp, u, j0);
  if ((threadIdx.x & 31) < 16) {
    int j = j0 + (threadIdx.x & 15);
    atomicAdd(err, fabsf(v[j] * t - 1.0f / (float)M_VOCAB));
  }
}

__global__ void sk_err_zero(float* __restrict__ err) { *err = 0.0f; }
__global__ void sk_err_set(const float* __restrict__ err, int* __restrict__ done) {
  if (*err <= THR) *done = 1;
}

// ---------------------------------------------------------------- u = (1/n)/(K v + eps)
template<int USEK>
__global__ void __launch_bounds__(256)
sk_rowdot_u(const float* __restrict__ Kp, const float* __restrict__ Mp,
            const float* __restrict__ v, float* __restrict__ u,
            const int* __restrict__ done) {
  if (*done) return;
  __shared__ float red[256];
  const int i   = blockIdx.x;
  const int tid = threadIdx.x;
  const long long row = (long long)i * M_VOCAB;
  float acc = 0.0f;
  for (int j = tid * 4; j < M_VOCAB; j += 256 * 4) {
    float4 kv;
    if (USEK) {
      kv = *(const float4*)(Kp + row + j);
    } else {
      float4 mv = *(const float4*)(Mp + row + j);
      kv.x = __expf(-ALPHA * mv.x); kv.y = __expf(-ALPHA * mv.y);
      kv.z = __expf(-ALPHA * mv.z); kv.w = __expf(-ALPHA * mv.w);
    }
    float4 vv = *(const float4*)(v + j);
    acc += kv.x * vv.x + kv.y * vv.y + kv.z * vv.z + kv.w * vv.w;
  }
  red[tid] = acc;
  __syncthreads();
  for (int s = 128; s > 0; s >>= 1) {
    if (tid < s) red[tid] += red[tid + s];
    __syncthreads();
  }
  if (tid == 0) u[i] = (1.0f / (float)N_TOPICS) / (red[0] + EPS_S);
}

// ---------------------------------------------------------------- loss = 100 * sum u_i K_ij M_ij v_j
__global__ void __launch_bounds__(256)
sk_loss(const float* __restrict__ Mp, const float* __restrict__ u,
        const float* __restrict__ v, float* __restrict__ out) {
  __shared__ float red[256];
  const int i   = blockIdx.x;
  const int tid = threadIdx.x;
  const long long row = (long long)i * M_VOCAB;
  float acc = 0.0f;
  for (int j = tid * 4; j < M_VOCAB; j += 256 * 4) {
    float4 mv = *(const float4*)(Mp + row + j);
    float4 vv = *(const float4*)(v + j);
    acc += mv.x * __expf(-ALPHA * mv.x) * vv.x;
    acc += mv.y * __expf(-ALPHA * mv.y) * vv.y;
    acc += mv.z * __expf(-ALPHA * mv.z) * vv.z;
    acc += mv.w * __expf(-ALPHA * mv.w) * vv.w;
  }
  red[tid] = acc;
  __syncthreads();
  for (int s = 128; s > 0; s >>= 1) {
    if (tid < s) red[tid] += red[tid + s];
    __syncthreads();
  }
  if (tid == 0) atomicAdd(out, u[i] * red[0] * WLOSS);
}

// ---------------------------------------------------------------- host
extern "C" void kernel_launch(void* const* d_in, const int* in_sizes, int n_in,
                              void* d_out, int out_size, void* d_ws, size_t ws_size,
                              hipStream_t stream) {
  (void)in_sizes; (void)n_in; (void)out_size;
  const float* M   = (const float*)d_in[0];
  float*       out = (float*)d_out;
  char*        ws  = (char*)d_ws;

  // Workspace layout (bytes):
  //   [0, 8000)        u  (2000 f32)
  //   [8000, 88000)    v  (20000 f32)
  //   [88064]          done (int)
  //   [88068]          err  (f32)
  //   [88320, ...)     K  (n*m f32) if it fits  -> L2-resident Gibbs kernel
  float* u    = (float*)(ws);
  float* v    = (float*)(ws + 8000);
  int*   done = (int*)  (ws + 88064);
  float* err  = (float*)(ws + 88068);
  float* K    = (float*)(ws + 88320);
  const size_t K_bytes = (size_t)N_TOPICS * M_VOCAB * sizeof(float); // 160 MB
  const bool useK = ws_size >= (size_t)88320 + K_bytes;

  sk_init<<<(N_TOPICS + 255) / 256, 256, 0, stream>>>(u, done, err, out);
  if (useK) {
    const long long q = (long long)N_TOPICS * M_VOCAB / 4;   // 10M float4s
    sk_buildK<<<(unsigned)((q + 255) / 256), 256, 0, stream>>>(M, K);
  }

  const unsigned colBlocks = M_VOCAB / 16;   // 1250 single-wave tiles
  for (int cpt = 1; cpt <= 100; ++cpt) {
    if (useK) sk_colsum_v<1><<<colBlocks, 32, 0, stream>>>(K, M, u, v, done);
    else      sk_colsum_v<0><<<colBlocks, 32, 0, stream>>>(K, M, u, v, done);

    if (useK) sk_rowdot_u<1><<<N_TOPICS, 256, 0, stream>>>(K, M, v, u, done);
    else      sk_rowdot_u<0><<<N_TOPICS, 256, 0, stream>>>(K, M, v, u, done);

    if (cpt == 1 || cpt == 51) {   // reference checks err only when cpt % 50 == 1
      sk_err_zero<<<1, 1, 0, stream>>>(err);
      if (useK) sk_colsum_err<1><<<colBlocks, 32, 0, stream>>>(K, M, u, v, err, done);
      else      sk_colsum_err<0><<<colBlocks, 32, 0, stream>>>(K, M, u, v, err, done);
      sk_err_set<<<1, 1, 0, stream>>>(err, done);
    }
  }

  sk_loss<<<N_TOPICS, 256, 0, stream>>>(M, u, v, out);
}